// ChrominanceBranch_75557064671630
// MI455X (gfx1250) — compile-verified
//
#include <hip/hip_runtime.h>

typedef __attribute__((ext_vector_type(2))) float v2f;
typedef __attribute__((ext_vector_type(8))) float v8f;

// Workspace layout (bytes):
//   [0,16384)      : uint32 hist [2 chan][32 img][64 bins]
//   [16384,17920)  : double  sums [2 chan][32 img][3] = {sum, sumsq, patchvar_sum}
//   [17920,34816)  : float   feats[32][132]
#define HIST_OFF  0
#define SUMS_OFF  16384
#define FEATS_OFF 17920
#define WS_WORDS  (34816 / 4)

__global__ void zero_ws_kernel(unsigned int* __restrict__ ws) {
  int i = blockIdx.x * blockDim.x + threadIdx.x;
  if (i < WS_WORDS) ws[i] = 0u;
}

// One block = one 8-row strip (8 x 1024) of one channel plane.
// grid = 32 img * 2 chan * 128 strips = 8192 blocks, 256 threads.
__global__ __launch_bounds__(256) void stats_pass_kernel(
    const float* __restrict__ ycbcr,
    unsigned int* __restrict__ g_hist,
    double* __restrict__ g_sums) {
  __shared__ unsigned int lhist[8][64];
  __shared__ float sh_sum, sh_sumsq, sh_pvar;

  const int tid  = threadIdx.x;
  const int wave = tid >> 5;

  for (int i = tid; i < 8 * 64; i += 256) ((unsigned int*)lhist)[i] = 0u;
  if (tid == 0) { sh_sum = 0.f; sh_sumsq = 0.f; sh_pvar = 0.f; }
  __syncthreads();

  const int strip = blockIdx.x & 127;
  const int ci    = blockIdx.x >> 7;   // ci = chan*32 + img, chan: 0=cb, 1=cr
  const int chan  = ci >> 5;
  const int img   = ci & 31;

  const float* base = ycbcr + ((size_t)(img * 3 + 1 + chan) << 20);
  const int row0 = strip << 3;

  unsigned int* myhist = lhist[wave];
  float s = 0.f, s2 = 0.f;  // thread covers 4 cols x 8 rows = half of one 8x8 patch

  for (int r = 0; r < 8; ++r) {
    const float4 v = *(const float4*)(base + (size_t)(row0 + r) * 1024 + tid * 4);
    float xs[4] = {v.x, v.y, v.z, v.w};
#pragma unroll
    for (int j = 0; j < 4; ++j) {
      float x = xs[j];
      int b = (int)(x * 0.25f);          // floor(x * 64/256), x >= 0
      b = b < 0 ? 0 : (b > 63 ? 63 : b);
      atomicAdd(&myhist[b], 1u);
      s  += x;
      s2 += x * x;
    }
  }

  // Combine lane pairs (t, t^1) -> one full 8x8 patch (n=64, unbiased var)
  float ps  = s  + __shfl_xor(s, 1, 32);
  float ps2 = s2 + __shfl_xor(s2, 1, 32);
  if ((tid & 1) == 0) {
    float pvar = (ps2 - ps * ps * (1.0f / 64.0f)) * (1.0f / 63.0f);
    atomicAdd(&sh_pvar, pvar);
  }
  atomicAdd(&sh_sum, s);
  atomicAdd(&sh_sumsq, s2);
  __syncthreads();

  if (tid < 64) {
    unsigned int t = 0;
#pragma unroll
    for (int w = 0; w < 8; ++w) t += lhist[w][tid];
    atomicAdd(&g_hist[ci * 64 + tid], t);
  }
  if (tid == 0) {
    atomicAdd(&g_sums[ci * 3 + 0], (double)sh_sum);
    atomicAdd(&g_sums[ci * 3 + 1], (double)sh_sumsq);
    atomicAdd(&g_sums[ci * 3 + 2], (double)sh_pvar);
  }
}

// grid = 32 (one block per image), 160 threads.
__global__ void finalize_feats_kernel(const unsigned int* __restrict__ g_hist,
                                      const double* __restrict__ g_sums,
                                      float* __restrict__ feats) {
  const int b = blockIdx.x;
  const int t = threadIdx.x;
  const double invN = 1.0 / (1048576.0 + 1e-8);
  if (t < 64) {
    feats[b * 132 + t] = (float)((double)g_hist[b * 64 + t] * invN);
  } else if (t < 128) {
    feats[b * 132 + t] = (float)((double)g_hist[(32 + b) * 64 + (t - 64)] * invN);
  } else if (t < 132) {
    int k = t - 128;
    int c = k >> 1;        // 0=cb, 1=cr
    int which = k & 1;     // 0=global var, 1=local var mean
    const double* sp = g_sums + (size_t)(c * 32 + b) * 3;
    double val;
    if (which == 0) {
      const double N = 1048576.0;
      val = (sp[1] - sp[0] * sp[0] / N) / (N - 1.0);
    } else {
      val = sp[2] / 16384.0;
    }
    feats[b * 132 + t] = (float)val;
  }
}

// Single block, 256 threads (8 waves). GEMM 32x256x132 via V_WMMA_F32_16X16X4_F32,
// then bias + ReLU + batchnorm over B=32.
__global__ __launch_bounds__(256) void mlp_bn_kernel(
    const float* __restrict__ feats_g,
    const float* __restrict__ W1,
    const float* __restrict__ b1,
    const float* __restrict__ gamma,
    const float* __restrict__ beta,
    float* __restrict__ out) {
  __shared__ float fA[32 * 132];
  __shared__ float hS[32 * 256];

  const int tid = threadIdx.x;
  for (int i = tid; i < 32 * 132; i += 256) fA[i] = feats_g[i];
  __syncthreads();  // EXEC all-ones from here through the WMMA section

  const int lane  = tid & 31;
  const int wave  = tid >> 5;
  const int row   = lane & 15;           // A row within tile
  const int koff  = (lane >> 4) << 1;    // lanes 0-15: K+{0,1}; lanes 16-31: K+{2,3}
  const int ncolL = lane & 15;           // B/C/D column within tile

  v8f c[2][2];
#pragma unroll
  for (int mt = 0; mt < 2; ++mt)
#pragma unroll
    for (int nt = 0; nt < 2; ++nt)
#pragma unroll
      for (int i = 0; i < 8; ++i) c[mt][nt][i] = 0.f;

  const int n0 = (wave * 2 + 0) * 16 + ncolL;
  const int n1 = (wave * 2 + 1) * 16 + ncolL;

#pragma unroll 1
  for (int kb = 0; kb < 132; kb += 4) {
    v2f a0, a1, b0, b1v;
    a0.x = fA[row * 132 + kb + koff];
    a0.y = fA[row * 132 + kb + koff + 1];
    a1.x = fA[(row + 16) * 132 + kb + koff];
    a1.y = fA[(row + 16) * 132 + kb + koff + 1];
    b0.x  = W1[(kb + koff) * 256 + n0];
    b0.y  = W1[(kb + koff + 1) * 256 + n0];
    b1v.x = W1[(kb + koff) * 256 + n1];
    b1v.y = W1[(kb + koff + 1) * 256 + n1];
    c[0][0] = __builtin_amdgcn_wmma_f32_16x16x4_f32(false, a0, false, b0,  (short)0, c[0][0], false, false);
    c[0][1] = __builtin_amdgcn_wmma_f32_16x16x4_f32(false, a0, false, b1v, (short)0, c[0][1], false, false);
    c[1][0] = __builtin_amdgcn_wmma_f32_16x16x4_f32(false, a1, false, b0,  (short)0, c[1][0], false, false);
    c[1][1] = __builtin_amdgcn_wmma_f32_16x16x4_f32(false, a1, false, b1v, (short)0, c[1][1], false, false);
  }

  // bias + ReLU, scatter C tiles (VGPR i -> row i or i+8) into hS
#pragma unroll
  for (int mt = 0; mt < 2; ++mt)
#pragma unroll
    for (int nt = 0; nt < 2; ++nt) {
      const int ncol = (wave * 2 + nt) * 16 + ncolL;
      const float bv = b1[ncol];
#pragma unroll
      for (int i = 0; i < 8; ++i) {
        const int r = mt * 16 + ((lane >> 4) << 3) + i;
        float v = c[mt][nt][i] + bv;
        hS[r * 256 + ncol] = v > 0.f ? v : 0.f;
      }
    }
  __syncthreads();

  // BatchNorm over batch axis: each thread owns one output column.
  {
    const int n = tid;
    float mu = 0.f;
#pragma unroll
    for (int b = 0; b < 32; ++b) mu += hS[b * 256 + n];
    mu *= (1.0f / 32.0f);
    float var = 0.f;
#pragma unroll
    for (int b = 0; b < 32; ++b) { float d = hS[b * 256 + n] - mu; var += d * d; }
    var *= (1.0f / 32.0f);
    const float inv = 1.0f / sqrtf(var + 1e-5f);
    const float g = gamma[n], be = beta[n];
#pragma unroll
    for (int b = 0; b < 32; ++b)
      out[b * 256 + n] = g * (hS[b * 256 + n] - mu) * inv + be;
  }
}

extern "C" void kernel_launch(void* const* d_in, const int* in_sizes, int n_in,
                              void* d_out, int out_size, void* d_ws, size_t ws_size,
                              hipStream_t stream) {
  const float* ycbcr = (const float*)d_in[0];
  const float* W1    = (const float*)d_in[1];
  const float* b1    = (const float*)d_in[2];
  const float* gamma = (const float*)d_in[3];
  const float* beta  = (const float*)d_in[4];
  float* out = (float*)d_out;

  unsigned int* ws_u32 = (unsigned int*)d_ws;
  unsigned int* g_hist = (unsigned int*)((char*)d_ws + HIST_OFF);
  double*       g_sums = (double*)((char*)d_ws + SUMS_OFF);
  float*        feats  = (float*)((char*)d_ws + FEATS_OFF);

  zero_ws_kernel<<<(WS_WORDS + 255) / 256, 256, 0, stream>>>(ws_u32);
  stats_pass_kernel<<<8192, 256, 0, stream>>>(ycbcr, g_hist, g_sums);
  finalize_feats_kernel<<<32, 160, 0, stream>>>(g_hist, g_sums, feats);
  mlp_bn_kernel<<<1, 256, 0, stream>>>(feats, W1, b1, gamma, beta, out);
}